// FDMAS_22204980920940
// MI455X (gfx1250) — compile-verified
//
#include <hip/hip_runtime.h>
#include <math.h>

// Problem sizes (fixed by the reference's setup_inputs)
#define NA 4        // angles
#define NE 128      // elements
#define NS 2048     // rf samples per trace
#define NZ 512      // depth samples (FFT axis)
#define NX 256      // lateral samples
#define NCOL (NA * NX)      // 1024 GEMM columns = (angle, x) pairs
#define NPIX (NZ * NX)      // 131072 = 2^17

typedef float v2f __attribute__((ext_vector_type(2)));
typedef float v8f __attribute__((ext_vector_type(8)));

// ---------------------------------------------------------------------------
// Kernel 1: DMAS beamform. One thread per (a,z,x) pixel; loop over elements.
// d_rx/apod reads are coalesced over x (stride-1), rf gathers hit L2 (rf=4MB).
// Writes fdmas into workspace as F[z][a*NX+x] (row = z) for the filter GEMM.
// ---------------------------------------------------------------------------
__global__ void fdmas_beamform(const float* __restrict__ rf,
                               const float* __restrict__ t0,
                               const float* __restrict__ d_tx,
                               const float* __restrict__ d_rx,
                               const float* __restrict__ apod,
                               const float* __restrict__ fs_p,
                               const float* __restrict__ c0_p,
                               float* __restrict__ F)
{
    int gid = blockIdx.x * blockDim.x + threadIdx.x;   // a*NZ*NX + z*NX + x
    if (gid >= NA * NZ * NX) return;
    int x   = gid & (NX - 1);
    int z   = (gid >> 8) & (NZ - 1);
    int a   = gid >> 17;                 // NPIX = 2^17
    int pix = gid & (NPIX - 1);          // z*NX + x

    const float fs    = fs_p[0];
    const float scale = fs / c0_p[0];    // (d/c0)*fs
    const float toff  = t0[a] * fs;
    const float dtx   = d_tx[gid];

    float sh = 0.0f;   // sum of sign(s)*sqrt(|s|)
    float sa = 0.0f;   // sum of |s|

    const float* rf_a = rf + (size_t)a * NE * NS;
    for (int e = 0; e < NE; ++e) {
        float dr = d_rx[e * NPIX + pix];
        float ap = apod[e * NPIX + pix];
        float delay = (dtx + dr) * scale - toff;
        float fl = floorf(delay);
        float w  = delay - fl;           // fractional weight
        int   i0 = (int)fl;
        const float* tr = rf_a + e * NS;
        float g0 = (i0 >= 0  && i0 < NS)     ? tr[i0]     : 0.0f; // zero padding
        float g1 = (i0 >= -1 && i0 < NS - 1) ? tr[i0 + 1] : 0.0f;
        float s  = (g0 * (1.0f - w) + g1 * w) * ap;
        float av = fabsf(s);
        sh += copysignf(sqrtf(av), s);
        sa += av;
    }
    F[z * NCOL + a * NX + x] = 0.5f * (sh * sh - sa);
}

// ---------------------------------------------------------------------------
// Kernel 2: impulse response h = irfft(H_tukey, n=512).
// H is real => filtering == circular convolution with h.
// h[n] = (1/N)*(H[0] + (-1)^n*H[N/2] + 2*sum_{k=1}^{N/2-1} H[k] cos(2*pi*k*n/N))
// Exact phase reduction (k*n mod N) keeps cospif args small & accurate.
// Launch: <<<1, NZ>>>.
// ---------------------------------------------------------------------------
__global__ void build_h(const float* __restrict__ fs_p,
                        const float* __restrict__ f0_p,
                        float* __restrict__ h)
{
    __shared__ float Hs[NZ / 2 + 1];
    const float fs = fs_p[0];
    const float f0 = f0_p[0];
    const float f1 = 2.0f * f0 - 0.7f * f0;   // BW = 0.7, DMAS band around 2*f0
    const float f2 = 2.0f * f0 + 0.7f * f0;

    int n = threadIdx.x;
    if (n <= NZ / 2) {
        float f = (float)n * (fs / (float)NZ);
        float u = (f - f1) / (f2 - f1);
        float w;
        if (u < 0.0f || u > 1.0f) {
            w = 0.0f;                                   // outside band
        } else if (u < 0.25f) {                         // ALPHA/2 = 0.25
            w = 0.5f * (1.0f + cospif(4.0f * u - 1.0f));            // rising taper
        } else if (u > 0.75f) {
            w = 0.5f * (1.0f + cospif(4.0f * (1.0f - u) - 1.0f));   // falling taper
        } else {
            w = 1.0f;
        }
        Hs[n] = w;
    }
    __syncthreads();

    float sum = Hs[0] + ((n & 1) ? -Hs[NZ / 2] : Hs[NZ / 2]);
    for (int k = 1; k < NZ / 2; ++k) {
        int m = (k * n) & (NZ - 1);                     // exact mod-N phase
        sum += 2.0f * Hs[k] * cospif((float)m * (2.0f / (float)NZ));
    }
    h[n] = sum * (1.0f / (float)NZ);
}

// ---------------------------------------------------------------------------
// Kernel 3: circulant filter as fp32 WMMA GEMM.
// Out(512 x 1024) = M(512 x 512) * F(512 x 1024), M[i][j] = h[(i-j) & 511].
// One wave32 per 16x16 tile, 128 steps of V_WMMA_F32_16X16X4_F32.
// A 16x4 f32 layout: lanes 0-15 hold K=kk+0,kk+1 (vgpr0,1); lanes 16-31 K=kk+2,kk+3.
// B 4x16 layout: vgpr0 = rows K=kk+{0|2}, vgpr1 = rows K=kk+{1|3} (half per lane group).
// C/D: vgpr v -> M = v + 8*(lane>=16), N = lane&15.
// EXEC is all ones (no divergence) as WMMA requires.
// ---------------------------------------------------------------------------
__global__ void filter_gemm(const float* __restrict__ F,
                            const float* __restrict__ h,
                            float* __restrict__ out)
{
    int lane = threadIdx.x & 31;
    int wave = (blockIdx.x * blockDim.x + threadIdx.x) >> 5;
    int tm   = wave >> 6;          // 32 tiles along z (M)
    int tn   = wave & 63;          // 64 tiles along columns (N)
    int half = lane >> 4;          // 0: lanes 0-15, 1: lanes 16-31
    int l15  = lane & 15;
    int row  = tm * 16 + l15;      // A-matrix row handled by this lane
    int col  = tn * 16 + l15;      // B/D column handled by this lane

    v8f acc = {};
    for (int kk = 0; kk < NZ; kk += 4) {
        int k0 = kk + half * 2;
        v2f aa, bb;
        aa.x = h[(row - k0)     & (NZ - 1)];   // M[row][k0]   (2KB table, cache-hot)
        aa.y = h[(row - k0 - 1) & (NZ - 1)];   // M[row][k0+1]
        bb.x = F[k0 * NCOL + col];             // coalesced row loads
        bb.y = F[(k0 + 1) * NCOL + col];
        acc = __builtin_amdgcn_wmma_f32_16x16x4_f32(
            /*neg_a=*/false, aa, /*neg_b=*/false, bb,
            /*c_mod=*/(short)0, acc, /*reuse_a=*/false, /*reuse_b=*/false);
    }

    int a  = col >> 8;             // angle = col / NX
    int xx = col & (NX - 1);
#pragma unroll
    for (int v = 0; v < 8; ++v) {
        int zz = tm * 16 + v + half * 8;
        out[a * NPIX + zz * NX + xx] = acc[v];
    }
}

// ---------------------------------------------------------------------------
extern "C" void kernel_launch(void* const* d_in, const int* in_sizes, int n_in,
                              void* d_out, int out_size, void* d_ws, size_t ws_size,
                              hipStream_t stream)
{
    (void)in_sizes; (void)n_in; (void)out_size; (void)ws_size;
    const float* rf   = (const float*)d_in[0];
    const float* t0   = (const float*)d_in[1];
    const float* d_tx = (const float*)d_in[2];
    const float* d_rx = (const float*)d_in[3];
    const float* fs   = (const float*)d_in[4];
    const float* f0   = (const float*)d_in[5];
    const float* c0   = (const float*)d_in[6];
    const float* apod = (const float*)d_in[7];
    float* out = (float*)d_out;

    float* F = (float*)d_ws;          // 512*1024 floats = 2 MB
    float* h = F + NZ * NCOL;         // 512 floats

    fdmas_beamform<<<(NA * NZ * NX + 255) / 256, 256, 0, stream>>>(
        rf, t0, d_tx, d_rx, apod, fs, c0, F);
    build_h<<<1, NZ, 0, stream>>>(fs, f0, h);
    // 32*64 = 2048 tiles, 8 waves (256 threads) per block -> 256 blocks
    filter_gemm<<<256, 256, 0, stream>>>(F, h, out);
}